// WaveletKANLayer_29996051595936
// MI455X (gfx1250) — compile-verified
//
#include <hip/hip_runtime.h>

typedef __attribute__((ext_vector_type(2))) float v2f;
typedef __attribute__((ext_vector_type(8))) float v8f;

#define F_DIM   256   // feature / K dimension
#define OUT_DIM 256   // output columns

// ---------------------------------------------------------------------------
// Prep: fold wavelet linear + biases into a single GEMM weight/bias.
//   Weff[o,f] = W_out[o,f] * s*(W_nl[f,0]+W_nl[f,1])
//   beff[o]   = b_out[o] + sum_f b_nl[f]*W_out[o,f]
// One block per output row o; 256 threads (one per f).
// ---------------------------------------------------------------------------
__global__ __launch_bounds__(F_DIM)
void wkan_prep_kernel(const float* __restrict__ W_nl,
                      const float* __restrict__ b_nl,
                      const float* __restrict__ W_out,
                      const float* __restrict__ b_out,
                      float* __restrict__ Weff,
                      float* __restrict__ beff) {
    __shared__ float red[F_DIM];
    const int o = blockIdx.x;
    const int f = threadIdx.x;
    const float s = 0.70710678118654752440f; // 1/sqrt(2)

    const float c = s * (W_nl[2 * f] + W_nl[2 * f + 1]);
    const float w = W_out[(size_t)o * F_DIM + f];
    Weff[(size_t)o * F_DIM + f] = w * c;

    red[f] = w * b_nl[f];
    __syncthreads();
#pragma unroll
    for (int stride = F_DIM / 2; stride > 0; stride >>= 1) {
        if (f < stride) red[f] += red[f + stride];
        __syncthreads();
    }
    if (f == 0) beff[o] = red[0] + b_out[o];
}

// ---------------------------------------------------------------------------
// GEMM: Out[M,256] = X[M,256] @ Weff^T + beff  via V_WMMA_F32_16X16X4_F32.
// Block = 256 threads (8 waves). Block tile = 32 rows x 256 cols:
//   wave -> rowg = wave>>2 (0..1), colg = wave&3 (0..3); wave tile 16x64.
// A fragment (16x4 f32): lane l holds row (l&15), K = kbase + (l>>4)*2 + {0,1}.
// B fragment (4x16 f32): lane l holds col (l&15), K = kbase + (l>>4)*2 + {0,1}.
// C/D (16x16 f32): VGPR r, lane l -> row = r + 8*(l>>4), col = (l&15).
// ---------------------------------------------------------------------------
__global__ __launch_bounds__(256, 2)
void wkan_gemm_kernel(const float* __restrict__ X,
                      const float* __restrict__ Weff,
                      const float* __restrict__ beff,
                      float* __restrict__ Out) {
    const int lane = threadIdx.x & 31;
    const int wave = threadIdx.x >> 5;
    const int rowg = wave >> 2;  // 0..1
    const int colg = wave & 3;   // 0..3

    const int row_base = blockIdx.x * 32 + rowg * 16;
    const int col_base = colg * 64;

    const int m15   = lane & 15;
    const int khalf = (lane >> 4) * 2;  // 0 or 2

    // Per-lane base pointers (consecutive K is contiguous -> b64 loads)
    const float* ap  = X    + (size_t)(row_base + m15) * F_DIM + khalf;
    const float* bp0 = Weff + (size_t)(col_base +  0 + m15) * F_DIM + khalf;
    const float* bp1 = Weff + (size_t)(col_base + 16 + m15) * F_DIM + khalf;
    const float* bp2 = Weff + (size_t)(col_base + 32 + m15) * F_DIM + khalf;
    const float* bp3 = Weff + (size_t)(col_base + 48 + m15) * F_DIM + khalf;

    v8f acc0 = {}, acc1 = {}, acc2 = {}, acc3 = {};

#pragma unroll 8
    for (int k = 0; k < F_DIM; k += 4) {
        v2f a  = *(const v2f*)(ap  + k);
        v2f b0 = *(const v2f*)(bp0 + k);
        v2f b1 = *(const v2f*)(bp1 + k);
        v2f b2 = *(const v2f*)(bp2 + k);
        v2f b3 = *(const v2f*)(bp3 + k);

        acc0 = __builtin_amdgcn_wmma_f32_16x16x4_f32(false, a, false, b0,
                                                     (short)0, acc0, false, false);
        acc1 = __builtin_amdgcn_wmma_f32_16x16x4_f32(false, a, false, b1,
                                                     (short)0, acc1, false, false);
        acc2 = __builtin_amdgcn_wmma_f32_16x16x4_f32(false, a, false, b2,
                                                     (short)0, acc2, false, false);
        acc3 = __builtin_amdgcn_wmma_f32_16x16x4_f32(false, a, false, b3,
                                                     (short)0, acc3, false, false);
    }

    // Epilogue: add per-column bias, store. Row for VGPR r: r + 8*(lane>>4).
    const int row0 = row_base + ((lane >> 4) << 3);

    const int c0 = col_base +  0 + m15;
    const int c1 = col_base + 16 + m15;
    const int c2 = col_base + 32 + m15;
    const int c3 = col_base + 48 + m15;
    const float bias0 = beff[c0];
    const float bias1 = beff[c1];
    const float bias2 = beff[c2];
    const float bias3 = beff[c3];

#pragma unroll
    for (int r = 0; r < 8; ++r) {
        const size_t rowoff = (size_t)(row0 + r) * OUT_DIM;
        Out[rowoff + c0] = acc0[r] + bias0;
        Out[rowoff + c1] = acc1[r] + bias1;
        Out[rowoff + c2] = acc2[r] + bias2;
        Out[rowoff + c3] = acc3[r] + bias3;
    }
}

// ---------------------------------------------------------------------------
extern "C" void kernel_launch(void* const* d_in, const int* in_sizes, int n_in,
                              void* d_out, int out_size, void* d_ws, size_t ws_size,
                              hipStream_t stream) {
    const float* x     = (const float*)d_in[0];  // [B, N, F] = [M, 256]
    const float* W_nl  = (const float*)d_in[1];  // [F, 2]
    const float* b_nl  = (const float*)d_in[2];  // [F]
    const float* W_out = (const float*)d_in[3];  // [OUT, F]
    const float* b_out = (const float*)d_in[4];  // [OUT]
    float* out = (float*)d_out;                  // [M, OUT]

    float* Weff = (float*)d_ws;                  // 256*256 floats
    float* beff = Weff + (size_t)OUT_DIM * F_DIM;

    const int M = in_sizes[0] / F_DIM;           // 8*20000 = 160000

    wkan_prep_kernel<<<OUT_DIM, F_DIM, 0, stream>>>(W_nl, b_nl, W_out, b_out,
                                                    Weff, beff);
    // M is divisible by 32 (160000 / 32 = 5000): no tail handling needed.
    wkan_gemm_kernel<<<M / 32, 256, 0, stream>>>(x, Weff, beff, out);
}